// Encoder_11536282157710
// MI455X (gfx1250) — compile-verified
//
#include <hip/hip_runtime.h>
#include <hip/hip_bf16.h>
#include <stdint.h>

typedef float v2f __attribute__((ext_vector_type(2)));
typedef float v8f __attribute__((ext_vector_type(8)));

#define IN_C 128
#define HID 64
#define WPAD 80   // LDS dword stride for W rows: 80 % 64 == 16 -> conflict-free B reads

// Async global->LDS copy of 8B per lane (one wave copies 256B = one W row).
// First operand = LDS byte address (low 32 bits of generic LDS pointer),
// second = 64-bit global address. Tracked by ASYNCcnt.
__device__ __forceinline__ void async_copy_b64(uint32_t lds_off, const void* gaddr) {
    asm volatile("global_load_async_to_lds_b64 %0, %1, off"
                 :: "v"(lds_off), "v"(gaddr) : "memory");
}
__device__ __forceinline__ void wait_async_zero() {
    asm volatile("s_wait_asynccnt 0x0" ::: "memory");
}

// ---------------- degree kernels ----------------
__global__ __launch_bounds__(256) void deg_init(float* __restrict__ deg, int n) {
    int i = blockIdx.x * blockDim.x + threadIdx.x;
    if (i < n) deg[i] = 1.0f;  // self loop
}

__global__ __launch_bounds__(256) void deg_count(const int* __restrict__ dst,
                                                 float* __restrict__ deg, int E) {
    int e = blockIdx.x * blockDim.x + threadIdx.x;
    if (e < E) unsafeAtomicAdd(&deg[dst[e]], 1.0f);
}

__global__ __launch_bounds__(256) void deg_rsqrt(float* __restrict__ deg, int n) {
    int i = blockIdx.x * blockDim.x + threadIdx.x;
    if (i < n) deg[i] = rsqrtf(deg[i]);
}

// ---------------- GEMM: h = x @ W, out = h * dinv^2 (self-loop seed) ----------------
// One wave computes a 16-row x 64-col tile with V_WMMA_F32_16X16X4_F32.
// A (16x4 f32): lanes 0-15 = rows, v0/v1 hold K=k0,k0+1 (lo half) or k0+2,k0+3 (hi half).
// B (4x16 f32): v0 = K row (k0 lo / k0+2 hi), v1 = next K row, N striped across lanes.
__global__ __launch_bounds__(256) void gcn_gemm(const float* __restrict__ x,
                                                const float* __restrict__ W,
                                                const float* __restrict__ dinv,
                                                float* __restrict__ h,
                                                float* __restrict__ out,
                                                int ntiles) {
    __shared__ float Ws[IN_C * WPAD];  // 40 KB, padded stride

    const int wid  = threadIdx.x >> 5;
    const int lane = threadIdx.x & 31;

    // Each of the 8 waves async-copies 16 rows of W (one row = 256B = wave32 x b64).
    {
        const uint32_t ldsBase = (uint32_t)(uintptr_t)&Ws[0];
#pragma unroll
        for (int r = 0; r < 16; ++r) {
            const int row = wid * 16 + r;
            async_copy_b64(ldsBase + (uint32_t)(row * WPAD + lane * 2) * 4u,
                           W + (size_t)row * HID + lane * 2);
        }
        wait_async_zero();
    }
    __syncthreads();

    const int tile = blockIdx.x * 8 + wid;
    if (tile >= ntiles) return;          // whole wave exits together -> EXEC all-1s for WMMA

    const bool hi  = lane >= 16;
    const int l    = lane & 15;
    const int rowBase = tile * 16;
    const int koff = hi ? 2 : 0;

    v8f acc0 = {}, acc1 = {}, acc2 = {}, acc3 = {};

    const float* xr = x + (size_t)(rowBase + l) * IN_C + koff;  // lane streams its own row
    const float* wB = Ws + koff * WPAD + l;

#pragma unroll 2
    for (int k0 = 0; k0 < IN_C; k0 += 4) {
        float2 av = *reinterpret_cast<const float2*>(xr + k0);
        v2f a; a.x = av.x; a.y = av.y;

        const float* wk = wB + k0 * WPAD;
        v2f b0, b1, b2, b3;
        b0.x = wk[0];   b0.y = wk[WPAD + 0];
        b1.x = wk[16];  b1.y = wk[WPAD + 16];
        b2.x = wk[32];  b2.y = wk[WPAD + 32];
        b3.x = wk[48];  b3.y = wk[WPAD + 48];

        acc0 = __builtin_amdgcn_wmma_f32_16x16x4_f32(false, a, false, b0, (short)0, acc0, false, false);
        acc1 = __builtin_amdgcn_wmma_f32_16x16x4_f32(false, a, false, b1, (short)0, acc1, false, false);
        acc2 = __builtin_amdgcn_wmma_f32_16x16x4_f32(false, a, false, b2, (short)0, acc2, false, false);
        acc3 = __builtin_amdgcn_wmma_f32_16x16x4_f32(false, a, false, b3, (short)0, acc3, false, false);
    }

    // C/D layout: VGPR r, lanes 0-15 -> M=r, N=lane; lanes 16-31 -> M=r+8, N=lane-16.
#pragma unroll
    for (int r = 0; r < 8; ++r) {
        const int row = rowBase + r + (hi ? 8 : 0);
        const float dv = dinv[row];
        const float s  = dv * dv;
        const size_t o = (size_t)row * HID + l;
        const float v0 = acc0[r], v1 = acc1[r], v2 = acc2[r], v3 = acc3[r];
        h[o]        = v0;  h[o + 16]   = v1;  h[o + 32]   = v2;  h[o + 48]   = v3;
        out[o]      = v0 * s;
        out[o + 16] = v1 * s;
        out[o + 32] = v2 * s;
        out[o + 48] = v3 * s;
    }
}

// ---------------- edge scatter: out[dst] += h[src] * dinv[src]*dinv[dst] ----------------
__global__ __launch_bounds__(256) void gcn_scatter(const int* __restrict__ src,
                                                   const int* __restrict__ dst,
                                                   const float* __restrict__ dinv,
                                                   const float* __restrict__ h,
                                                   float* __restrict__ out, int E) {
    int t = blockIdx.x * blockDim.x + threadIdx.x;
    int e = t >> 4;
    if (e >= E) return;
    int q = t & 15;
    int s = src[e];
    int d = dst[e];
    float w = dinv[s] * dinv[d];
    float4 hv = *reinterpret_cast<const float4*>(h + (size_t)s * HID + q * 4);
    float* op = out + (size_t)d * HID + q * 4;
    unsafeAtomicAdd(op + 0, hv.x * w);
    unsafeAtomicAdd(op + 1, hv.y * w);
    unsafeAtomicAdd(op + 2, hv.z * w);
    unsafeAtomicAdd(op + 3, hv.w * w);
}

// ---------------- bias + PReLU ----------------
__global__ __launch_bounds__(256) void gcn_epilogue(float* __restrict__ out,
                                                    const float* __restrict__ b,
                                                    const float* __restrict__ pw,
                                                    int total4) {
    int i = blockIdx.x * blockDim.x + threadIdx.x;
    if (i >= total4) return;
    int c4 = i & 15;  // 16 float4 groups per row of 64
    float4 v  = reinterpret_cast<float4*>(out)[i];
    float4 bb = reinterpret_cast<const float4*>(b)[c4];
    float4 pp = reinterpret_cast<const float4*>(pw)[c4];
    v.x += bb.x; v.y += bb.y; v.z += bb.z; v.w += bb.w;
    v.x = v.x >= 0.0f ? v.x : pp.x * v.x;
    v.y = v.y >= 0.0f ? v.y : pp.y * v.y;
    v.z = v.z >= 0.0f ? v.z : pp.z * v.z;
    v.w = v.w >= 0.0f ? v.w : pp.w * v.w;
    reinterpret_cast<float4*>(out)[i] = v;
}

extern "C" void kernel_launch(void* const* d_in, const int* in_sizes, int n_in,
                              void* d_out, int out_size, void* d_ws, size_t ws_size,
                              hipStream_t stream) {
    const float* x  = (const float*)d_in[0];
    const int*   ei = (const int*)d_in[1];    // edge_index [2, E] row-major
    const float* W  = (const float*)d_in[2];
    const float* b  = (const float*)d_in[3];
    const float* pw = (const float*)d_in[4];
    float* out = (float*)d_out;

    const int N = in_sizes[0] / IN_C;
    const int E = in_sizes[1] / 2;
    const int* src = ei;
    const int* dst = ei + E;

    // workspace layout: dinv[N] | h[N*HID]
    float* dinv = (float*)d_ws;
    size_t hOff = ((size_t)N * sizeof(float) + 255) & ~(size_t)255;
    float* h = (float*)((char*)d_ws + hOff);

    deg_init<<<(N + 255) / 256, 256, 0, stream>>>(dinv, N);
    deg_count<<<(E + 255) / 256, 256, 0, stream>>>(dst, dinv, E);
    deg_rsqrt<<<(N + 255) / 256, 256, 0, stream>>>(dinv, N);

    const int ntiles = (N + 15) / 16;            // 6250
    gcn_gemm<<<(ntiles + 7) / 8, 256, 0, stream>>>(x, W, dinv, h, out, ntiles);

    long long sthreads = (long long)E * 16;      // 25.6M
    gcn_scatter<<<(int)((sthreads + 255) / 256), 256, 0, stream>>>(src, dst, dinv, h, out, E);

    const int total4 = N * (HID / 4);
    gcn_epilogue<<<(total4 + 255) / 256, 256, 0, stream>>>(out, b, pw, total4);
}